// QuantumEnhancedQLSTM_65481071400708
// MI455X (gfx1250) — compile-verified
//
#include <hip/hip_runtime.h>
#include <math.h>

typedef __attribute__((ext_vector_type(2))) float v2f;
typedef __attribute__((ext_vector_type(8))) float v8f;

#define TT 512
#define BB 1024
#define DD 128
#define HH 4
#define GROWS 16  // 4 gates * 4 rows each

// ------------------------------------------------------------------
// Pack Wf/Wi/Wu/Wo [4,132] into Wcat[16,128] (x-part), Wh[16,4]
// (h-part), bias[16]. Row order: gate-major (f,i,u,o), row-minor.
// ------------------------------------------------------------------
__global__ void qlstm_pack_kernel(const float* __restrict__ Wf, const float* __restrict__ Wi,
                                  const float* __restrict__ Wu, const float* __restrict__ Wo,
                                  const float* __restrict__ bf, const float* __restrict__ bi,
                                  const float* __restrict__ bu, const float* __restrict__ bo,
                                  float* __restrict__ Wcat, float* __restrict__ Wh,
                                  float* __restrict__ bias) {
  int idx = blockIdx.x * blockDim.x + threadIdx.x;
  if (idx < GROWS * (DD + HH)) {
    int row = idx / (DD + HH);
    int col = idx % (DD + HH);
    int gate = row >> 2, r = row & 3;
    const float* W = (gate == 0) ? Wf : (gate == 1) ? Wi : (gate == 2) ? Wu : Wo;
    float v = W[r * (DD + HH) + col];
    if (col < DD) Wcat[row * DD + col] = v;
    else          Wh[row * HH + (col - DD)] = v;
  }
  if (idx < GROWS) {
    int gate = idx >> 2, r = idx & 3;
    const float* bsrc = (gate == 0) ? bf : (gate == 1) ? bi : (gate == 2) ? bu : bo;
    bias[idx] = bsrc[r];
  }
}

// ------------------------------------------------------------------
// GEMM: out[T*B,16] = A[T*B,128] * Wcat[16,128]^T via
// V_WMMA_F32_16X16X4_F32 (exact fp32, K=4/op, 32 chained per tile).
// One wave per 16-row tile; 8 waves per block. Zero-reuse stream:
// direct global->VGPR feeding WMMA (no LDS/TDM round-trip needed).
// A frag: lanes 0-15 -> (M=lane, K={k,k+1}), lanes 16-31 -> K={k+2,k+3}.
// B frag: transpose-symmetric layout => same load pattern from Wcat.
// ------------------------------------------------------------------
__global__ void __launch_bounds__(256)
qlstm_gemm_wmma_kernel(const float* __restrict__ A, const float* __restrict__ Wcat,
                       float* __restrict__ out) {
  const int lane = threadIdx.x & 31;
  const int wave = threadIdx.x >> 5;
  const long tile = (long)blockIdx.x * 8 + wave;
  const long row0 = tile * 16;
  const int r = lane & 15;
  const int koff = (lane >> 4) * 2;
  const float* arow = A + (row0 + r) * DD;
  const float* brow = Wcat + r * DD;

  v8f c = {0.f, 0.f, 0.f, 0.f, 0.f, 0.f, 0.f, 0.f};
#pragma unroll
  for (int k = 0; k < DD; k += 4) {
    v2f a = *reinterpret_cast<const v2f*>(arow + k + koff);
    v2f b = *reinterpret_cast<const v2f*>(brow + k + koff);
    c = __builtin_amdgcn_wmma_f32_16x16x4_f32(false, a, false, b, (short)0, c, false, false);
  }

  const int mbase = (lane >> 4) * 8;
  const int n = lane & 15;
#pragma unroll
  for (int j = 0; j < 8; ++j) {
    out[(row0 + mbase + j) * 16 + n] = c[j];  // each half-wave: 64B contiguous per row
  }
}

// ------------------------------------------------------------------
// 4-qubit statevector, amplitudes as float2 (re,im) so the RX pair
// update z0' = c*z0 + s*(-i*z1) lowers to packed VOP3P f32 math.
// Wire w <-> flat-index bit (3-w).
// ------------------------------------------------------------------
__device__ __forceinline__ void rx_apply(v2f z[16], int bit, float c, float s) {
#pragma unroll
  for (int base = 0; base < 16; ++base) {
    if (base & (1 << bit)) continue;
    const int i1 = base | (1 << bit);
    v2f z0 = z[base], z1 = z[i1];
    v2f j1 = {z1.y, -z1.x};  // -i * z1
    v2f j0 = {z0.y, -z0.x};  // -i * z0
    z[base] = c * z0 + s * j1;
    z[i1]   = c * z1 + s * j0;
  }
}

__device__ __forceinline__ void cnot_apply(v2f z[16], int bc, int bt) {
#pragma unroll
  for (int i = 0; i < 16; ++i) {
    if ((i & (1 << bc)) && !(i & (1 << bt))) {
      const int j = i | (1 << bt);
      v2f t = z[i]; z[i] = z[j]; z[j] = t;
    }
  }
}

__device__ __forceinline__ void qsim(const float pre[4], const float ct[8], const float st[8],
                                     float ez[4]) {
  v2f z[16];
#pragma unroll
  for (int i = 0; i < 16; ++i) z[i] = (v2f){0.f, 0.f};
  z[0] = (v2f){1.f, 0.f};
  // RX encoding (per-element angles)
#pragma unroll
  for (int w = 0; w < 4; ++w) {
    float h = 0.5f * pre[w];
    rx_apply(z, 3 - w, __cosf(h), __sinf(h));
  }
  // DEPTH x (fixed RX layer + CNOT chain); CNOTs are register renames
#pragma unroll
  for (int d = 0; d < 2; ++d) {
#pragma unroll
    for (int w = 0; w < 4; ++w) rx_apply(z, 3 - w, ct[d * 4 + w], st[d * 4 + w]);
#pragma unroll
    for (int w = 0; w < 3; ++w) cnot_apply(z, 3 - w, 3 - (w + 1));
  }
  // <Z_w>
#pragma unroll
  for (int w = 0; w < 4; ++w) ez[w] = 0.f;
#pragma unroll
  for (int i = 0; i < 16; ++i) {
    float p = fmaf(z[i].x, z[i].x, z[i].y * z[i].y);
#pragma unroll
    for (int w = 0; w < 4; ++w) ez[w] += ((i >> (3 - w)) & 1) ? -p : p;
  }
}

// Fast activations on the serial chain: v_exp_f32 + v_rcp_f32.
__device__ __forceinline__ float fast_sigmoid(float x) {
  return __fdividef(1.f, 1.f + __expf(-x));
}
__device__ __forceinline__ float fast_tanh(float x) {
  return fmaf(2.f, __fdividef(1.f, 1.f + __expf(-2.f * x)), -1.f);
}

// ------------------------------------------------------------------
// Sequential scan. One lane per (batch, gate): quad of lanes per
// batch element. h/c + statevector live in VGPRs for the whole T
// loop; gate activations exchanged within the quad via __shfl
// (ds_bpermute, wave32 lockstep -> no barriers). Next step's gate
// row is prefetched into a register before the qsim chain.
// ------------------------------------------------------------------
__global__ void __launch_bounds__(64)
qlstm_scan_kernel(const float* __restrict__ gates, const float* __restrict__ Wh,
                  const float* __restrict__ bias, const float* __restrict__ thetas,
                  const float* __restrict__ h0, const float* __restrict__ c0,
                  float* __restrict__ outs, float* __restrict__ hxf, float* __restrict__ cxf) {
  const int g = blockIdx.x * blockDim.x + threadIdx.x;  // 0..4095
  const int b = g >> 2;                                  // batch element
  const int q = g & 3;                                   // gate: 0=f 1=i 2=u 3=o
  const int lane = threadIdx.x & 31;
  const int qbase = lane & ~3;

  float whx[4][4];
#pragma unroll
  for (int j = 0; j < 4; ++j)
#pragma unroll
    for (int k = 0; k < 4; ++k) whx[j][k] = Wh[(q * 4 + j) * 4 + k];

  float bq[4];
#pragma unroll
  for (int j = 0; j < 4; ++j) bq[j] = bias[q * 4 + j];

  float ct[8], st[8];
#pragma unroll
  for (int i = 0; i < 8; ++i) {
    float h = 0.5f * thetas[i];
    ct[i] = __cosf(h);
    st[i] = __sinf(h);
  }

  float hx[4], cx[4];
#pragma unroll
  for (int j = 0; j < 4; ++j) { hx[j] = h0[b * 4 + j]; cx[j] = c0[b * 4 + j]; }

  // q==2 (update gate) -> tanh, others -> sigmoid; tanh(x)=2*sigmoid(2x)-1
  const float insc = (q == 2) ? 2.f : 1.f;
  const float mm   = (q == 2) ? 2.f : 1.f;
  const float aa   = (q == 2) ? -1.f : 0.f;

  const float* gbase = gates + (size_t)b * 16 + q * 4;
  float4 gx = *reinterpret_cast<const float4*>(gbase);

  for (int t = 0; t < TT; ++t) {
    // prefetch next step's gate row (clamped select, stays in workspace)
    const int tn = (t + 1 < TT) ? t + 1 : t;
    float4 gx_next = *reinterpret_cast<const float4*>(gbase + (size_t)tn * BB * 16);

    float pre[4] = {gx.x + bq[0], gx.y + bq[1], gx.z + bq[2], gx.w + bq[3]};
#pragma unroll
    for (int j = 0; j < 4; ++j)
#pragma unroll
      for (int k = 0; k < 4; ++k) pre[j] = fmaf(hx[k], whx[j][k], pre[j]);

    float ez[4];
    qsim(pre, ct, st, ez);

    float act[4];
#pragma unroll
    for (int j = 0; j < 4; ++j) {
      float ss = fast_sigmoid(ez[j] * insc);
      act[j] = fmaf(ss, mm, aa);
    }

#pragma unroll
    for (int j = 0; j < 4; ++j) {
      float fj = __shfl(act[j], qbase + 0, 32);
      float ij = __shfl(act[j], qbase + 1, 32);
      float uj = __shfl(act[j], qbase + 2, 32);
      float oj = __shfl(act[j], qbase + 3, 32);
      cx[j] = fmaf(fj, cx[j], ij * uj);
      hx[j] = oj * fast_tanh(cx[j]);
    }
    float hq = (q == 0) ? hx[0] : (q == 1) ? hx[1] : (q == 2) ? hx[2] : hx[3];
    outs[((size_t)t * BB + b) * 4 + q] = hq;
    gx = gx_next;
  }
  float hq = (q == 0) ? hx[0] : (q == 1) ? hx[1] : (q == 2) ? hx[2] : hx[3];
  float cq = (q == 0) ? cx[0] : (q == 1) ? cx[1] : (q == 2) ? cx[2] : cx[3];
  hxf[b * 4 + q] = hq;
  cxf[b * 4 + q] = cq;
}

// ------------------------------------------------------------------
extern "C" void kernel_launch(void* const* d_in, const int* in_sizes, int n_in,
                              void* d_out, int out_size, void* d_ws, size_t ws_size,
                              hipStream_t stream) {
  const float* inputs = (const float*)d_in[0];   // [T,B,D]
  const float* Wf     = (const float*)d_in[1];
  const float* bf     = (const float*)d_in[2];
  const float* Wi     = (const float*)d_in[3];
  const float* bi     = (const float*)d_in[4];
  const float* Wu     = (const float*)d_in[5];
  const float* bu     = (const float*)d_in[6];
  const float* Wo     = (const float*)d_in[7];
  const float* bo     = (const float*)d_in[8];
  const float* thetas = (const float*)d_in[9];   // [8]
  const float* h0     = (const float*)d_in[10];  // [B,4]
  const float* c0     = (const float*)d_in[11];  // [B,4]

  float* wsf    = (float*)d_ws;
  float* gatesx = wsf;                                  // T*B*16 fp32 (32 MB)
  float* Wcat   = gatesx + (size_t)TT * BB * 16;        // 16*128
  float* Whbuf  = Wcat + GROWS * DD;                    // 16*4
  float* biasb  = Whbuf + GROWS * HH;                   // 16

  float* outs = (float*)d_out;                 // [T,B,4]
  float* hxf  = outs + (size_t)TT * BB * HH;   // [B,4]
  float* cxf  = hxf + (size_t)BB * HH;         // [B,4]

  qlstm_pack_kernel<<<9, 256, 0, stream>>>(Wf, Wi, Wu, Wo, bf, bi, bu, bo,
                                           Wcat, Whbuf, biasb);

  const int tiles = (TT * BB) / 16;  // 32768, 8 waves/block
  qlstm_gemm_wmma_kernel<<<tiles / 8, 256, 0, stream>>>(inputs, Wcat, gatesx);

  qlstm_scan_kernel<<<(BB * 4) / 64, 64, 0, stream>>>(gatesx, Whbuf, biasb, thetas,
                                                      h0, c0, outs, hxf, cxf);
}